// EosLayer_71725953843449
// MI455X (gfx1250) — compile-verified
//
#include <hip/hip_runtime.h>

typedef __attribute__((ext_vector_type(2))) float v2f;
typedef __attribute__((ext_vector_type(8))) float v8f;

#define DIM    512
#define KEXP   8
#define NPACK  544           // 512 (i) + 8 (e) + 8 (s) + 16 zero pad  (divisible by 32)
#define NUSED  528
#define BATCH  4
#define SEQ    4096
#define ROWS   (BATCH*SEQ)   // 16384
#define TAU    16.0f
#define LN_EPS 1e-5f
#define CHUNK  128
#define NCHUNK (SEQ/CHUNK)   // 32
#define NBLK   32            // GEMM block/wave tile width in N

__device__ __forceinline__ float log_sigmoid(float x) {
    return (x >= 0.f) ? -log1pf(__expf(-x)) : (x - log1pf(__expf(x)));
}

// ---------------------------------------------------------------------------
// Pack W = [W_i | W_e | W_s | 0pad]  (512 x 544), row-major (in, out)
// ---------------------------------------------------------------------------
__global__ void pack_w(const float* __restrict__ Wi, const float* __restrict__ We,
                       const float* __restrict__ Ws, float* __restrict__ Wp) {
    int idx = blockIdx.x * blockDim.x + threadIdx.x;
    if (idx >= DIM * NPACK) return;
    int k = idx / NPACK, n = idx % NPACK;
    float v;
    if (n < DIM)            v = Wi[k * DIM + n];
    else if (n < DIM + 8)   v = We[k * 8 + (n - DIM)];
    else if (n < NUSED)     v = Ws[k * 8 + (n - DIM - 8)];
    else                    v = 0.f;
    Wp[idx] = v;
}

// ---------------------------------------------------------------------------
// C[M x N] = A[M x 512] * B[512 x N]   (fp32 exact, V_WMMA_F32_16X16X4_F32)
// Block: 256 threads = 8 waves; block tile 128 rows x 32 cols.
// Wave tile 16 x 32: one A fragment feeds 2 WMMAs (2 N-subtiles).
// B panel in LDS with K-pair interleaving: element (k,n) at
//   Bs[(k/2)*NBLK*2 + n*2 + (k&1)]
// so B fragments are ds_load_(2addr_)b64 with offsets folded into the
// instruction. 4-deep K software pipeline; last outer iteration peeled.
// `#pragma unroll 1` on the main loop keeps a single accumulator tuple
// (WMMA D aliases C in place -> no v_mov shuffles).
// ---------------------------------------------------------------------------
__global__ __launch_bounds__(256) void gemm_f32_wmma(const float* __restrict__ A,
                                                     const float* __restrict__ B,
                                                     float* __restrict__ C, int N) {
    __shared__ float Bs[DIM * NBLK];               // 64 KB
    const int col0   = blockIdx.x * NBLK;
    const int rowBlk = blockIdx.y;

    // cooperative staging with K-pair interleave (global reads coalesced float4)
    for (int i = threadIdx.x; i < DIM * NBLK / 4; i += 256) {
        int k  = i / (NBLK / 4);                   // 0..511
        int n4 = (i % (NBLK / 4)) * 4;
        float4 v = *(const float4*)(B + (size_t)k * N + col0 + n4);
        float* dst = &Bs[(size_t)(k >> 1) * NBLK * 2 + (k & 1)];
        dst[(n4 + 0) * 2] = v.x;
        dst[(n4 + 1) * 2] = v.y;
        dst[(n4 + 2) * 2] = v.z;
        dst[(n4 + 3) * 2] = v.w;
    }
    __syncthreads();

    const int wave = threadIdx.x >> 5;             // 0..7  (M sub-block)
    const int lane = threadIdx.x & 31;
    const int half = lane >> 4;                    // 0: K0/K1, 1: K2/K3
    const int ln16 = lane & 15;
    const int row  = rowBlk * 128 + wave * 16 + ln16;
    const float* __restrict__ Aptr = A + (size_t)row * DIM + 2 * half;
    // B fragment base: pair index 'half', column ln16 (and +16 for subtile 1)
    const float* __restrict__ Bp0 = &Bs[((size_t)half * NBLK + ln16) * 2];
    const float* __restrict__ Bp1 = Bp0 + 32;

    v2f aF[4], b0F[4], b1F[4];                     // 4-deep fragment pipeline
    auto loadStage = [&](int s, int k) {
        aF[s]  = *(const v2f*)(Aptr + k);                       // global_load_b64
        b0F[s] = *(const v2f*)(Bp0 + (size_t)k * NBLK);         // ds_load_b64
        b1F[s] = *(const v2f*)(Bp1 + (size_t)k * NBLK);         // ds_load_b64
    };

#pragma unroll
    for (int s = 0; s < 4; ++s) loadStage(s, s * 4);

    v8f acc0 = {0.f, 0.f, 0.f, 0.f, 0.f, 0.f, 0.f, 0.f};
    v8f acc1 = {0.f, 0.f, 0.f, 0.f, 0.f, 0.f, 0.f, 0.f};

#pragma unroll 1
    for (int k = 0; k < DIM - 16; k += 16) {       // main loop: loads always in range
        __builtin_prefetch(Aptr + k + 128, 0, 1);  // speculative, no branch
#pragma unroll
        for (int u = 0; u < 4; ++u) {
            acc0 = __builtin_amdgcn_wmma_f32_16x16x4_f32(
                false, aF[u], false, b0F[u], (short)0, acc0, false, false);
            acc1 = __builtin_amdgcn_wmma_f32_16x16x4_f32(
                false, aF[u], false, b1F[u], (short)0, acc1, false, false);
            loadStage(u, k + 4 * u + 16);
        }
    }
#pragma unroll
    for (int u = 0; u < 4; ++u) {                  // peeled epilogue: K = 496..508
        acc0 = __builtin_amdgcn_wmma_f32_16x16x4_f32(
            false, aF[u], false, b0F[u], (short)0, acc0, false, false);
        acc1 = __builtin_amdgcn_wmma_f32_16x16x4_f32(
            false, aF[u], false, b1F[u], (short)0, acc1, false, false);
    }

    // D layout: lanes 0-15 -> M = j, lanes 16-31 -> M = 8 + j, N = ln16
    const int rowBase = rowBlk * 128 + wave * 16 + half * 8;
#pragma unroll
    for (int j = 0; j < 8; ++j) {
        C[(size_t)(rowBase + j) * N + col0 + ln16]      = acc0[j];
        C[(size_t)(rowBase + j) * N + col0 + 16 + ln16] = acc1[j];
    }
}

// ---------------------------------------------------------------------------
// Scan pass A: per (batch, chunk, d) compute local chunk-end state (zero init).
// state layout: [b*NCHUNK][KEXP][DIM]
// ---------------------------------------------------------------------------
__global__ __launch_bounds__(512) void scan_chunk_local(const float* __restrict__ IES,
                                                        const float* __restrict__ o_param,
                                                        float* __restrict__ state) {
    const int bc = blockIdx.x;
    const int batch = bc / NCHUNK, chunk = bc % NCHUNK;
    const int d = threadIdx.x;

    float o[KEXP], m[KEXP];
#pragma unroll
    for (int k = 0; k < KEXP; ++k) {
        o[k] = __expf(log_sigmoid(o_param[k * DIM + d]) / TAU);
        m[k] = 0.f;
    }
    const int t0 = batch * SEQ + chunk * CHUNK;
    for (int t = 0; t < CHUNK; ++t) {
        const float* __restrict__ rowp = IES + (size_t)(t0 + t) * NPACK;
        float iv = rowp[d];
#pragma unroll
        for (int k = 0; k < KEXP; ++k)
            m[k] = o[k] * m[k] + rowp[DIM + k] * iv;   // e_t[k] * i_t[d]
    }
    float* __restrict__ sp = state + (size_t)bc * KEXP * DIM + d;
#pragma unroll
    for (int k = 0; k < KEXP; ++k) sp[k * DIM] = m[k];
}

// ---------------------------------------------------------------------------
// Scan pass B: prefix over chunks; decay^CHUNK in closed form from o_param.
// ---------------------------------------------------------------------------
__global__ __launch_bounds__(512) void scan_chunk_prefix(const float* __restrict__ state,
                                                         const float* __restrict__ o_param,
                                                         float* __restrict__ mstart) {
    const int batch = blockIdx.x;
    const int d = threadIdx.x;
    float oL[KEXP], m[KEXP];
#pragma unroll
    for (int k = 0; k < KEXP; ++k) {
        oL[k] = __expf((float)CHUNK * log_sigmoid(o_param[k * DIM + d]) / TAU);
        m[k] = 0.f;
    }
    for (int c = 0; c < NCHUNK; ++c) {
        size_t base = (size_t)(batch * NCHUNK + c) * KEXP * DIM + d;
#pragma unroll
        for (int k = 0; k < KEXP; ++k) {
            mstart[base + k * DIM] = m[k];
            m[k] = oL[k] * m[k] + state[base + k * DIM];
        }
    }
}

// ---------------------------------------------------------------------------
// Scan pass C: replay each chunk from its prefix state, emit y.
// ---------------------------------------------------------------------------
__global__ __launch_bounds__(512) void scan_emit(const float* __restrict__ IES,
                                                 const float* __restrict__ o_param,
                                                 const float* __restrict__ mstart,
                                                 float* __restrict__ Y) {
    const int bc = blockIdx.x;
    const int batch = bc / NCHUNK, chunk = bc % NCHUNK;
    const int d = threadIdx.x;

    float o[KEXP], m[KEXP];
    size_t base = (size_t)bc * KEXP * DIM + d;
#pragma unroll
    for (int k = 0; k < KEXP; ++k) {
        o[k] = __expf(log_sigmoid(o_param[k * DIM + d]) / TAU);
        m[k] = mstart[base + k * DIM];
    }
    const int t0 = batch * SEQ + chunk * CHUNK;
    for (int t = 0; t < CHUNK; ++t) {
        const float* __restrict__ rowp = IES + (size_t)(t0 + t) * NPACK;
        float iv = rowp[d];
        float y = 0.f;
#pragma unroll
        for (int k = 0; k < KEXP; ++k) {
            m[k] = o[k] * m[k] + rowp[DIM + k] * iv;
            y += rowp[DIM + 8 + k] * m[k];             // s_t[k] * m[k][d]
        }
        Y[(size_t)(t0 + t) * DIM + d] = y;
    }
}

// ---------------------------------------------------------------------------
// In-place LayerNorm over rows of 512 (block = 256 threads, 2 elems/thread).
// ---------------------------------------------------------------------------
__global__ __launch_bounds__(256) void layernorm_inplace(float* __restrict__ Y,
                                                         const float* __restrict__ gamma,
                                                         const float* __restrict__ beta) {
    __shared__ float red[256];
    float* __restrict__ yp = Y + (size_t)blockIdx.x * DIM;
    const int t = threadIdx.x;
    float v0 = yp[t], v1 = yp[t + 256];
    red[t] = v0 + v1;
    __syncthreads();
    for (int s = 128; s > 0; s >>= 1) {
        if (t < s) red[t] += red[t + s];
        __syncthreads();
    }
    float mu = red[0] * (1.f / DIM);
    __syncthreads();
    float d0 = v0 - mu, d1 = v1 - mu;
    red[t] = d0 * d0 + d1 * d1;
    __syncthreads();
    for (int s = 128; s > 0; s >>= 1) {
        if (t < s) red[t] += red[t + s];
        __syncthreads();
    }
    float rinv = rsqrtf(red[0] * (1.f / DIM) + LN_EPS);
    yp[t]       = d0 * rinv * gamma[t]       + beta[t];
    yp[t + 256] = d1 * rinv * gamma[t + 256] + beta[t + 256];
}

// ---------------------------------------------------------------------------
extern "C" void kernel_launch(void* const* d_in, const int* in_sizes, int n_in,
                              void* d_out, int out_size, void* d_ws, size_t ws_size,
                              hipStream_t stream) {
    (void)in_sizes; (void)n_in; (void)out_size; (void)ws_size;
    const float* x     = (const float*)d_in[0];
    const float* W_i   = (const float*)d_in[1];
    const float* W_e   = (const float*)d_in[2];
    const float* W_s   = (const float*)d_in[3];
    const float* o_p   = (const float*)d_in[4];
    const float* gamma = (const float*)d_in[5];
    const float* beta  = (const float*)d_in[6];
    const float* W_out = (const float*)d_in[7];
    float* out = (float*)d_out;

    float* ws     = (float*)d_ws;
    size_t off    = 0;
    float* Wp     = ws + off; off += (size_t)DIM * NPACK;                  // 1.1 MB
    float* IES    = ws + off; off += (size_t)ROWS * NPACK;                 // 35.7 MB
    float* Y      = ws + off; off += (size_t)ROWS * DIM;                   // 33.5 MB
    float* state  = ws + off; off += (size_t)BATCH * NCHUNK * KEXP * DIM;  // 2 MB
    float* mstart = ws + off; off += (size_t)BATCH * NCHUNK * KEXP * DIM;  // 2 MB

    pack_w<<<(DIM * NPACK + 255) / 256, 256, 0, stream>>>(W_i, W_e, W_s, Wp);

    dim3 g1(NPACK / NBLK, ROWS / 128);
    gemm_f32_wmma<<<g1, 256, 0, stream>>>(x, Wp, IES, NPACK);

    scan_chunk_local <<<BATCH * NCHUNK, 512, 0, stream>>>(IES, o_p, state);
    scan_chunk_prefix<<<BATCH,          512, 0, stream>>>(state, o_p, mstart);
    scan_emit        <<<BATCH * NCHUNK, 512, 0, stream>>>(IES, o_p, mstart, Y);

    layernorm_inplace<<<ROWS, 256, 0, stream>>>(Y, gamma, beta);

    dim3 g2(DIM / NBLK, ROWS / 128);
    gemm_f32_wmma<<<g2, 256, 0, stream>>>(Y, W_out, out, DIM);
}